// SAGEBlock_90761248899605
// MI455X (gfx1250) — compile-verified
//
#include <hip/hip_runtime.h>
#include <cstdint>

// ---------------------------------------------------------------------------
// GraphSAGE 3-layer forward for MI455X (gfx1250, wave32, WMMA).
//   layer: Y = act( mean_agg(X) @ Wl + X @ Wr + b )
// fp32 kept end-to-end; GEMMs use V_WMMA_F32_16X16X4_F32.
// ---------------------------------------------------------------------------

typedef float v2f __attribute__((ext_vector_type(2)));
typedef float v8f __attribute__((ext_vector_type(8)));

// ---------------- degree / reciprocal-degree ------------------------------
__global__ void deg_kernel(const long long* __restrict__ dst,
                           float* __restrict__ deg, int E) {
  int i = blockIdx.x * blockDim.x + threadIdx.x;
  if (i < E) atomicAdd(&deg[(int)dst[i]], 1.0f);
}

__global__ void rdeg_kernel(const float* __restrict__ deg,
                            float* __restrict__ rdeg, int N) {
  int i = blockIdx.x * blockDim.x + threadIdx.x;
  if (i < N) rdeg[i] = 1.0f / fmaxf(deg[i], 1.0f);
}

// ---------------- edge scatter (segment-sum into AGG) ---------------------
// One thread per (edge, feature). Consecutive threads cover consecutive
// features of the same edge -> coalesced gather of the src row and coalesced
// 64B bursts of L2 atomics into the dst row. AGG (<=51MB) lives in L2.
template <int F>
__global__ void scatter_kernel(const float* __restrict__ X,
                               const long long* __restrict__ src,
                               const long long* __restrict__ dst,
                               float* __restrict__ AGG, long long total) {
  long long i = (long long)blockIdx.x * blockDim.x + threadIdx.x;
  if (i >= total) return;
  int e = (int)(i / F);       // F is a power of two -> shifts
  int f = (int)(i % F);
  int s = (int)src[e];
  int d = (int)dst[e];
  atomicAdd(&AGG[(long long)d * F + f], X[(long long)s * F + f]);
}

// ---------------- fused SAGE GEMM: Y = act(AGG*rdeg @ Wl + X @ Wr + b) ----
// One wave computes a 16-row x 64-col output strip with 4 WMMA accumulators.
// A-fragment (16x4 f32, ISA layout): lane<16 holds {K=k0,k0+1} of row lane,
// lane>=16 holds {K=k0+2,k0+3} of row lane-16  -> one aligned b64 load.
// B-fragment (4x16): v0 = W[k0+2*half][col], v1 = W[k0+2*half+1][col].
// C/D (16x16 f32): VGPR r, lane L -> M = r + 8*(L>>4), N = L&15.
template <int K, int FO, int ACT>   // ACT: 0=relu, 1=sigmoid
__global__ __launch_bounds__(128)
void sage_gemm_kernel(const float* __restrict__ AGG,   // [M,K]
                      const float* __restrict__ X,     // [M,K]
                      const float* __restrict__ Wl,    // [K,FO]
                      const float* __restrict__ Wr,    // [K,FO]
                      const float* __restrict__ bias,  // [FO]
                      const float* __restrict__ rdeg,  // [M]
                      float* __restrict__ Y,           // [M,FO]
                      int M) {
  constexpr int NT     = 4;               // 4 x 16 = 64 output cols per wave
  constexpr int STRIPS = FO / (16 * NT);  // col strips per 16-row tile

  const int lane      = threadIdx.x & 31;
  const int waveInBlk = threadIdx.x >> 5;
  const int gw        = blockIdx.x * 4 + waveInBlk;
  const int rowTile   = gw / STRIPS;
  const int strip     = gw % STRIPS;
  const int rowTiles  = M >> 4;           // M = 50000 = 3125 * 16
  if (rowTile >= rowTiles) return;        // uniform per wave -> EXEC stays full

  const int rowInTile = lane & 15;
  const int kHalf     = lane >> 4;        // which K pair this lane holds
  const int row       = (rowTile << 4) + rowInTile;
  const int colBase   = strip * (16 * NT);
  const float rd      = rdeg[row];        // row fixed per lane: load once

  v8f acc[NT] = {};

  const float* arow = AGG + (long long)row * K + 2 * kHalf;
  const float* xrow = X   + (long long)row * K + 2 * kHalf;

  // ---- pass 1: (AGG * rdeg) @ Wl  (mean folded into A-fragment) ----
#pragma unroll 4
  for (int k0 = 0; k0 < K; k0 += 4) {
    v2f a = *(const v2f*)(arow + k0);     // 8B-aligned b64 load
    a *= rd;
    const float* wk = Wl + (long long)(k0 + 2 * kHalf) * FO + colBase + rowInTile;
#pragma unroll
    for (int t = 0; t < NT; ++t) {
      v2f b;
      b.x = wk[t * 16];
      b.y = wk[t * 16 + FO];
      acc[t] = __builtin_amdgcn_wmma_f32_16x16x4_f32(
          false, a, false, b, (short)0, acc[t], false, false);
    }
  }

  // ---- pass 2: X @ Wr ----
#pragma unroll 4
  for (int k0 = 0; k0 < K; k0 += 4) {
    v2f a = *(const v2f*)(xrow + k0);
    const float* wk = Wr + (long long)(k0 + 2 * kHalf) * FO + colBase + rowInTile;
#pragma unroll
    for (int t = 0; t < NT; ++t) {
      v2f b;
      b.x = wk[t * 16];
      b.y = wk[t * 16 + FO];
      acc[t] = __builtin_amdgcn_wmma_f32_16x16x4_f32(
          false, a, false, b, (short)0, acc[t], false, false);
    }
  }

  // ---- epilogue: + bias, activation, store ----
  const int outRowBase = (rowTile << 4) + (kHalf << 3);
#pragma unroll
  for (int t = 0; t < NT; ++t) {
    const int col = colBase + t * 16 + rowInTile;
    const float bv = bias[col];
#pragma unroll
    for (int r = 0; r < 8; ++r) {
      float v = acc[t][r] + bv;
      if (ACT == 0) v = fmaxf(v, 0.0f);
      else          v = 1.0f / (1.0f + __expf(-v));
      Y[(long long)(outRowBase + r) * FO + col] = v;
    }
  }
}

// ---------------------------------------------------------------------------
extern "C" void kernel_launch(void* const* d_in, const int* in_sizes, int n_in,
                              void* d_out, int out_size, void* d_ws, size_t ws_size,
                              hipStream_t stream) {
  const float*     x   = (const float*)d_in[0];
  const long long* ei  = (const long long*)d_in[1];   // int64 edge_index [2,E]
  const float*     Wl1 = (const float*)d_in[2];
  const float*     Wr1 = (const float*)d_in[3];
  const float*     b1  = (const float*)d_in[4];
  const float*     Wl2 = (const float*)d_in[5];
  const float*     Wr2 = (const float*)d_in[6];
  const float*     b2  = (const float*)d_in[7];
  const float*     Wl3 = (const float*)d_in[8];
  const float*     Wr3 = (const float*)d_in[9];
  const float*     b3  = (const float*)d_in[10];
  float*           out = (float*)d_out;

  const int N = in_sizes[0] / 128;   // 50000
  const int E = in_sizes[1] / 2;     // 800000
  const long long* src = ei;
  const long long* dst = ei + E;

  // workspace layout (floats): deg | rdeg | AGG(N*256, reused) | H1(N*256) | H2(N*128)
  float* base = (float*)d_ws;
  float* deg  = base;
  float* rdeg = base + N;
  float* AGG  = base + 2LL * N;
  float* H1   = AGG + 256LL * N;
  float* H2   = H1 + 256LL * N;

  const int BS = 256;

  // degrees (once, shared by all layers)
  hipMemsetAsync(deg, 0, (size_t)N * sizeof(float), stream);
  deg_kernel<<<(E + BS - 1) / BS, BS, 0, stream>>>(dst, deg, E);
  rdeg_kernel<<<(N + BS - 1) / BS, BS, 0, stream>>>(deg, rdeg, N);

  // ---------------- layer 1: 128 -> 256, ReLU ----------------
  hipMemsetAsync(AGG, 0, (size_t)N * 128 * sizeof(float), stream);
  {
    long long total = (long long)E * 128;
    scatter_kernel<128><<<(int)((total + BS - 1) / BS), BS, 0, stream>>>(
        x, src, dst, AGG, total);
    int waves = (N / 16) * (256 / 64);
    sage_gemm_kernel<128, 256, 0><<<(waves + 3) / 4, 128, 0, stream>>>(
        AGG, x, Wl1, Wr1, b1, rdeg, H1, N);
  }

  // ---------------- layer 2: 256 -> 128, ReLU ----------------
  hipMemsetAsync(AGG, 0, (size_t)N * 256 * sizeof(float), stream);
  {
    long long total = (long long)E * 256;
    scatter_kernel<256><<<(int)((total + BS - 1) / BS), BS, 0, stream>>>(
        H1, src, dst, AGG, total);
    int waves = (N / 16) * (128 / 64);
    sage_gemm_kernel<256, 128, 0><<<(waves + 3) / 4, 128, 0, stream>>>(
        AGG, H1, Wl2, Wr2, b2, rdeg, H2, N);
  }

  // ---------------- layer 3: 128 -> 64, Sigmoid --------------
  hipMemsetAsync(AGG, 0, (size_t)N * 128 * sizeof(float), stream);
  {
    long long total = (long long)E * 128;
    scatter_kernel<128><<<(int)((total + BS - 1) / BS), BS, 0, stream>>>(
        H2, src, dst, AGG, total);
    int waves = (N / 16) * (64 / 64);
    sage_gemm_kernel<128, 64, 1><<<(waves + 3) / 4, 128, 0, stream>>>(
        AGG, H2, Wl3, Wr3, b3, rdeg, out, N);
  }
}